// BlockDiagonalLinearAlignment_77189152244257
// MI455X (gfx1250) — compile-verified
//
#include <hip/hip_runtime.h>

typedef float v2f __attribute__((ext_vector_type(2)));
typedef float v8f __attribute__((ext_vector_type(8)));

#define BATCH   262144
#define DIM     128
#define NBLK    8
#define BSZ     16
#define RANK    4
#define EPS_F   1e-8f
#define WELEMS  (NBLK * BSZ * BSZ)   // 2048 floats = 8KB

// ---------------------------------------------------------------------------
// Kernel 1: fold dense / diagonal / lowrank blocks into one W[8][16][16]
// table (row-major per block, W[k][o][d]) so the main kernel is a single
// uniform block-diagonal GEMM.  Block pattern: k%3==0 dense, ==1 diag, ==2 LR.
// ---------------------------------------------------------------------------
__global__ void bdl_build_weights(const float* __restrict__ Wd,   // [3,16,16]
                                  const float* __restrict__ sd,   // [3,16]
                                  const float* __restrict__ U,    // [2,16,4]
                                  const float* __restrict__ V,    // [2,16,4]
                                  float* __restrict__ W)          // [8,16,16]
{
    int t = blockIdx.x * blockDim.x + threadIdx.x;
    if (t >= WELEMS) return;
    int k = t >> 8;          // block id 0..7
    int o = (t >> 4) & 15;   // out index
    int d = t & 15;          // in index
    int m = k % 3;
    float w;
    if (m == 0) {                         // dense blocks 0,3,6
        w = Wd[(k / 3) * 256 + o * 16 + d];
    } else if (m == 1) {                  // diagonal blocks 1,4,7
        w = (o == d) ? sd[((k - 1) / 3) * 16 + o] : 0.0f;
    } else {                              // lowrank blocks 2,5:  W = U * V^T
        int i = (k - 2) / 3;
        float acc = 0.0f;
#pragma unroll
        for (int r = 0; r < RANK; ++r)
            acc += U[i * 64 + o * 4 + r] * V[i * 64 + d * 4 + r];
        w = acc;
    }
    W[t] = w;
}

// ---------------------------------------------------------------------------
// Kernel 2: y = x @ blockdiag(W)^T, fused row-wise L2 normalization.
// One wave per 16-row tile; per block k: 4 chained V_WMMA_F32_16X16X4_F32.
//
// A (16x4 f32) layout: lanes 0-15 hold M=lane, VGPR0/1 = K=0,1;
//                      lanes 16-31 hold M=lane-16, VGPR0/1 = K=2,3.
//   -> per-lane float2 load from x at column (k*16 + 4c + 2*hi).
// B (4x16 f32) layout (mirror): lanes 0-15 hold N=lane, K=0,1;
//                               lanes 16-31 hold N=lane-16, K=2,3.
//   B[d,n] = W[n,d] -> per-lane float2 from row-major W[k][n][4c + 2*hi].
// C/D (16x16 f32): VGPR v holds row (v + 8*hi), lane column (lane&15).
// ---------------------------------------------------------------------------
__global__ __launch_bounds__(256) void bdl_main(const float* __restrict__ x,
                                                const float* __restrict__ W,
                                                float* __restrict__ out)
{
    __shared__ float lW[WELEMS];
    for (int i = threadIdx.x; i < WELEMS; i += 256)
        lW[i] = W[i];
    __syncthreads();

    const int lane = threadIdx.x & 31;
    const int wave = threadIdx.x >> 5;
    const int rlo  = lane & 15;   // A row / B col / D col
    const int hi   = lane >> 4;   // half-wave selector

    const long tile    = (long)blockIdx.x * 8 + wave;  // 16 rows per tile
    const long rowBase = tile * BSZ;
    const float* xrow  = x + (rowBase + rlo) * DIM;

    v8f acc[NBLK];
#pragma unroll
    for (int k = 0; k < NBLK; ++k) {
        v8f c = {0.f, 0.f, 0.f, 0.f, 0.f, 0.f, 0.f, 0.f};
#pragma unroll
        for (int cc = 0; cc < 4; ++cc) {
            v2f a = *(const v2f*)(xrow + k * 16 + cc * 4 + 2 * hi);
            v2f b = *(const v2f*)(&lW[k * 256 + rlo * 16 + cc * 4 + 2 * hi]);
            c = __builtin_amdgcn_wmma_f32_16x16x4_f32(
                    /*neg_a=*/false, a, /*neg_b=*/false, b,
                    /*c_mod=*/(short)0, c, /*reuse_a=*/false, /*reuse_b=*/false);
        }
        acc[k] = c;
    }

    // Fused L2 norm: per-lane partial sum over blocks, then reduce across the
    // 16 lanes of each half (masks 1,2,4,8 never cross the half boundary).
    float inv[8];
#pragma unroll
    for (int v = 0; v < 8; ++v) {
        float p = 0.0f;
#pragma unroll
        for (int k = 0; k < NBLK; ++k) {
            float t = acc[k][v];
            p += t * t;
        }
        p += __shfl_xor(p, 1);
        p += __shfl_xor(p, 2);
        p += __shfl_xor(p, 4);
        p += __shfl_xor(p, 8);
        inv[v] = 1.0f / (sqrtf(p) + EPS_F);
    }

    // Scale + store: per (k,v) the wave writes two contiguous 64B row segments.
    float* obase = out + rowBase * DIM;
#pragma unroll
    for (int k = 0; k < NBLK; ++k) {
#pragma unroll
        for (int v = 0; v < 8; ++v) {
            int row = v + 8 * hi;
            obase[(long)row * DIM + k * 16 + rlo] = acc[k][v] * inv[v];
        }
    }
}

// ---------------------------------------------------------------------------
extern "C" void kernel_launch(void* const* d_in, const int* in_sizes, int n_in,
                              void* d_out, int out_size, void* d_ws, size_t ws_size,
                              hipStream_t stream)
{
    (void)in_sizes; (void)n_in; (void)out_size; (void)ws_size;
    const float* x  = (const float*)d_in[0];
    const float* Wd = (const float*)d_in[1];
    const float* sd = (const float*)d_in[2];
    const float* U  = (const float*)d_in[3];
    const float* V  = (const float*)d_in[4];
    float* out = (float*)d_out;
    float* W   = (float*)d_ws;   // 8KB effective block weights

    bdl_build_weights<<<(WELEMS + 255) / 256, 256, 0, stream>>>(Wd, sd, U, V, W);

    const int tiles = BATCH / BSZ;            // 16384 waves
    bdl_main<<<tiles / 8, 256, 0, stream>>>(x, W, out);
}